// MM_DFN_4037269259078
// MI455X (gfx1250) — compile-verified
//
#include <hip/hip_runtime.h>
#include <math.h>

typedef __attribute__((ext_vector_type(2))) float v2f;
typedef __attribute__((ext_vector_type(8))) float v8f;

namespace {

constexpr int B_   = 32;
constexpr int L_   = 64;
constexpr int H_   = 200;
constexpr int M_   = 3;
constexpr int A_   = B_ * L_;   // 2048 nodes per modality
constexpr int N_   = M_ * A_;   // 6144 total nodes
constexpr int TAG_ = 6;
constexpr float PI_F = 3.14159265358979323846f;

// ---------------------------------------------------------------------------
// One wave computes one 16x16 f32 tile of C = A * B via V_WMMA_F32_16X16X4_F32.
// ISA layouts (wave32):
//   A 16x4 : lane = 16*(k/2) + m, reg = k%2     -> lane loads A[m][kk],A[m][kk+1]
//   B 4x16 : lane = 16*(k/2) + n, reg = k%2     -> lane loads B[kk][n],B[kk+1][n]
//   C 16x16: m = v + 8*(lane/16), n = lane%16
// ---------------------------------------------------------------------------
template <typename FA, typename FB, typename FE>
__device__ __forceinline__ void wmma_tile_f32(int m0, int n0, int K,
                                              FA loadA, FB loadB, FE epi) {
  const int lane = threadIdx.x & 31;
  const int mr   = lane & 15;
  const int kh   = lane >> 4;
  v8f acc = {};
  for (int k = 0; k < K; k += 4) {
    const int kk = k + 2 * kh;
    v2f a, b;
    a.x = loadA(m0 + mr, kk);
    a.y = loadA(m0 + mr, kk + 1);
    b.x = loadB(kk, n0 + mr);
    b.y = loadB(kk + 1, n0 + mr);
    acc = __builtin_amdgcn_wmma_f32_16x16x4_f32(false, a, false, b,
                                                (short)0, acc, false, false);
  }
#pragma unroll
  for (int v = 0; v < 8; ++v) epi(m0 + 8 * kh + v, n0 + mr, acc[v]);
}

__device__ __forceinline__ float wave_sum(float s) {
#pragma unroll
  for (int off = 16; off > 0; off >>= 1) s += __shfl_xor(s, off, 32);
  return s;
}

// --- row L2-normalize + concatenate raw features -> big [N,H], nrm [N,H] ----
__global__ void k_norm(const float* __restrict__ xa, const float* __restrict__ xv,
                       const float* __restrict__ xt, float* __restrict__ big,
                       float* __restrict__ nrm) {
  const int row  = blockIdx.x * 8 + (threadIdx.x >> 5);  // 0..N-1
  const int lane = threadIdx.x & 31;
  const float* src = (row < A_)     ? xa + (size_t)row * H_
                   : (row < 2 * A_) ? xv + (size_t)(row - A_) * H_
                                    : xt + (size_t)(row - 2 * A_) * H_;
  float s = 0.f;
  for (int k = lane; k < H_; k += 32) { float v = src[k]; s += v * v; }
  s = wave_sum(s);
  const float rn = 1.0f / sqrtf(s);
  for (int k = lane; k < H_; k += 32) {
    const float v = src[k];
    big[(size_t)row * H_ + k] = v;
    nrm[(size_t)row * H_ + k] = v * rn;
  }
}

// --- 96 per-(modality,dialogue) 64x64 Gram blocks -> angular similarity -----
__global__ void k_gram(const float* __restrict__ nrm, float* __restrict__ intra) {
  const int mb = blockIdx.x;                  // m*32 + b, 0..95
  const int w  = threadIdx.x >> 5;            // 16 waves -> 4x4 tiles
  const int m0 = (w >> 2) * 16, n0 = (w & 3) * 16;
  const float* blk = nrm + (size_t)mb * L_ * H_;
  auto lA = [&](int i, int k) { return blk[(size_t)i * H_ + k]; };
  auto lB = [&](int k, int j) { return blk[(size_t)j * H_ + k]; };
  auto ep = [&](int i, int j, float v) {
    intra[((size_t)mb * L_ + i) * L_ + j] = 1.0f - acosf(v * 0.99999f) * (1.0f / PI_F);
  };
  wmma_tile_f32(m0, n0, H_, lA, lB, ep);
}

// --- cross-modal per-utterance angular similarity, 6 ordered pairs ----------
__global__ void k_cross(const float* __restrict__ nrm, float* __restrict__ crossv) {
  const int idx  = blockIdx.x * 8 + (threadIdx.x >> 5);  // 0..6*A_-1
  const int lane = threadIdx.x & 31;
  const int pm[6] = {0, 0, 1, 1, 2, 2};
  const int pn[6] = {1, 2, 0, 2, 0, 1};
  const int q = idx / A_, a = idx % A_;
  const int m = pm[q], n = pn[q];
  const float* pa = nrm + ((size_t)m * A_ + a) * H_;
  const float* pb = nrm + ((size_t)n * A_ + a) * H_;
  float s = 0.f;
  for (int k = lane; k < H_; k += 32) s += pa[k] * pb[k];
  s = wave_sum(s);
  if (lane == 0)
    crossv[((size_t)m * M_ + n) * A_ + a] = 1.0f - acosf(s * 0.99999f) * (1.0f / PI_F);
}

// --- degree^(-1/2) per node -------------------------------------------------
__global__ void k_dinv(const float* __restrict__ intra, const float* __restrict__ crossv,
                       float* __restrict__ dinv) {
  const int row  = blockIdx.x * 8 + (threadIdx.x >> 5);  // node id
  const int lane = threadIdx.x & 31;
  const int m = row >> 11, a = row & (A_ - 1);
  const float* r = intra + (size_t)row * L_;   // (mb*64+i)*64 == row*64
  float s = 0.f;
  for (int j = lane; j < L_; j += 32) s += r[j];
  s = wave_sum(s);
  if (lane == 0) {
#pragma unroll
    for (int n = 0; n < M_; ++n)
      if (n != m) s += crossv[((size_t)m * M_ + n) * A_ + a];
    dinv[row] = 1.0f / sqrtf(s);
  }
}

// --- h0 = relu(big @ fc0_w + fc0_b) ----------------------------------------
__global__ void k_fc0(const float* __restrict__ big, const float* __restrict__ w,
                      const float* __restrict__ bias, float* __restrict__ h0) {
  const int wv = threadIdx.x >> 5;
  const int m0 = (blockIdx.y * 8 + wv) * 16, n0 = blockIdx.x * 16;
  auto lA = [&](int m, int k) { return big[(size_t)m * H_ + k]; };
  auto lB = [&](int k, int n) { return (n < H_) ? w[(size_t)k * H_ + n] : 0.0f; };
  auto ep = [&](int m, int n, float v) {
    if (n < H_) h0[(size_t)m * H_ + n] = fmaxf(v + bias[n], 0.0f);
  };
  wmma_tile_f32(m0, n0, H_, lA, lB, ep);
}

// --- hi = (D^-1/2 A D^-1/2) @ h using block structure -----------------------
// per (m,b): 64x64 similarity block (cols pre-scaled by dinv) times h block,
// plus 2 cross-modal diagonal terms, then row scale by dinv.
__global__ void k_spmm(const float* __restrict__ intra, const float* __restrict__ crossv,
                       const float* __restrict__ dinv, const float* __restrict__ h,
                       float* __restrict__ hi) {
  const int mb = blockIdx.y;          // 0..95
  const int m  = mb >> 5;
  const int rowbase = mb * L_;        // global node id of block row 0
  const int m0 = (threadIdx.x >> 5) * 16;   // 4 waves cover 64 rows
  const int n0 = blockIdx.x * 16;
  const float* S = intra + (size_t)mb * L_ * L_;
  auto lA = [&](int i, int k) { return S[(size_t)i * L_ + k] * dinv[rowbase + k]; };
  auto lB = [&](int k, int n) { return (n < H_) ? h[(size_t)(rowbase + k) * H_ + n] : 0.0f; };
  auto ep = [&](int i, int n, float acc) {
    if (n >= H_) return;
    const int row = rowbase + i;
    const int a   = row - m * A_;
    float s = acc;
#pragma unroll
    for (int nn = 0; nn < M_; ++nn) {
      if (nn == m) continue;
      const int orow = nn * A_ + a;
      s += crossv[((size_t)m * M_ + nn) * A_ + a] * dinv[orow] * h[(size_t)orow * H_ + n];
    }
    hi[(size_t)row * H_ + n] = s * dinv[row];
  };
  wmma_tile_f32(m0, n0, L_, lA, lB, ep);
}

// --- GCNII layer: hnew = relu(beta*([hi|h0] @ W) + (1-beta)*(0.9*hi+0.1*h0))
__global__ void k_conv(const float* __restrict__ hi, const float* __restrict__ h0,
                       const float* __restrict__ w, float beta, float* __restrict__ hout) {
  const int wv = threadIdx.x >> 5;
  const int m0 = (blockIdx.y * 8 + wv) * 16, n0 = blockIdx.x * 16;
  auto lA = [&](int m, int k) {
    return (k < H_) ? hi[(size_t)m * H_ + k] : h0[(size_t)m * H_ + (k - H_)];
  };
  auto lB = [&](int k, int n) { return (n < H_) ? w[(size_t)k * H_ + n] : 0.0f; };
  auto ep = [&](int m, int n, float v) {
    if (n >= H_) return;
    const float r = 0.9f * hi[(size_t)m * H_ + n] + 0.1f * h0[(size_t)m * H_ + n];
    hout[(size_t)m * H_ + n] = fmaxf(beta * v + (1.0f - beta) * r, 0.0f);
  };
  wmma_tile_f32(m0, n0, 2 * H_, lA, lB, ep);
}

// --- hx[m] = relu([big_m | hfin_m] @ tm_w[m] + tm_b[m]) ---------------------
__global__ void k_hx(const float* __restrict__ big, const float* __restrict__ hf,
                     const float* __restrict__ tw, const float* __restrict__ tb,
                     float* __restrict__ hx) {
  const int m  = blockIdx.z;
  const int wv = threadIdx.x >> 5;
  const int a0 = (blockIdx.y * 8 + wv) * 16, n0 = blockIdx.x * 16;
  const float* W  = tw + (size_t)m * 2 * H_ * H_;
  const float* Bb = tb + (size_t)m * H_;
  const int rb = m * A_;
  auto lA = [&](int a, int k) {
    return (k < H_) ? big[(size_t)(rb + a) * H_ + k] : hf[(size_t)(rb + a) * H_ + (k - H_)];
  };
  auto lB = [&](int k, int n) { return (n < H_) ? W[(size_t)k * H_ + n] : 0.0f; };
  auto ep = [&](int a, int n, float v) {
    if (n < H_) hx[(size_t)(rb + a) * H_ + n] = fmaxf(v + Bb[n], 0.0f);
  };
  wmma_tile_f32(a0, n0, 2 * H_, lA, lB, ep);
}

// --- one gated-attention pair p=(i,j): hout[i] += z*hx_i + (1-z)*hx_j -------
__global__ void k_gate(const float* __restrict__ hx, const float* __restrict__ cw,
                       const float* __restrict__ cb, float* __restrict__ hout,
                       int p, int i, int j) {
  const int wv = threadIdx.x >> 5;
  const int a0 = (blockIdx.y * 8 + wv) * 16, n0 = blockIdx.x * 16;
  const float* Hi = hx + (size_t)i * A_ * H_;
  const float* Hj = hx + (size_t)j * A_ * H_;
  const float* W  = cw + (size_t)p * 3 * H_ * H_;
  const float* Bb = cb + (size_t)p * H_;
  auto lA = [&](int a, int k) {
    if (k < H_) return Hi[(size_t)a * H_ + k];
    if (k < 2 * H_) return Hj[(size_t)a * H_ + (k - H_)];
    const int kk = k - 2 * H_;
    return Hi[(size_t)a * H_ + kk] * Hj[(size_t)a * H_ + kk];
  };
  auto lB = [&](int k, int n) { return (n < H_) ? W[(size_t)k * H_ + n] : 0.0f; };
  auto ep = [&](int a, int n, float v) {
    if (n >= H_) return;
    const float z = 1.0f / (1.0f + expf(-(v + Bb[n])));
    hout[(size_t)(i * A_ + a) * H_ + n] +=
        z * Hi[(size_t)a * H_ + n] + (1.0f - z) * Hj[(size_t)a * H_ + n];
  };
  wmma_tile_f32(a0, n0, 3 * H_, lA, lB, ep);
}

// --- joint logits: out[a,t] = sum_m (hout_m[a] @ uni_w[m] + uni_b[m]) -------
__global__ void k_final(const float* __restrict__ hout, const float* __restrict__ uw,
                        const float* __restrict__ ub, float* __restrict__ out) {
  const int a = blockIdx.x * blockDim.x + threadIdx.x;  // 0..A_-1
  float acc[TAG_];
#pragma unroll
  for (int t = 0; t < TAG_; ++t)
    acc[t] = ub[t] + ub[TAG_ + t] + ub[2 * TAG_ + t];
  for (int m = 0; m < M_; ++m) {
    const float* hr = hout + (size_t)(m * A_ + a) * H_;
    const float* wm = uw + (size_t)m * H_ * TAG_;
    for (int h = 0; h < H_; ++h) {
      const float v = hr[h];
#pragma unroll
      for (int t = 0; t < TAG_; ++t) acc[t] += v * wm[h * TAG_ + t];
    }
  }
#pragma unroll
  for (int t = 0; t < TAG_; ++t) out[(size_t)a * TAG_ + t] = acc[t];
}

}  // namespace

extern "C" void kernel_launch(void* const* d_in, const int* in_sizes, int n_in,
                              void* d_out, int out_size, void* d_ws, size_t ws_size,
                              hipStream_t stream) {
  (void)in_sizes; (void)n_in; (void)out_size; (void)ws_size;
  const float* x_a     = (const float*)d_in[0];
  const float* x_v     = (const float*)d_in[1];
  const float* x_t     = (const float*)d_in[2];
  const float* fc0_w   = (const float*)d_in[3];
  const float* fc0_b   = (const float*)d_in[4];
  const float* conv_w  = (const float*)d_in[5];
  const float* tm_w    = (const float*)d_in[6];
  const float* tm_b    = (const float*)d_in[7];
  const float* cross_w = (const float*)d_in[8];
  const float* cross_b = (const float*)d_in[9];
  const float* uni_w   = (const float*)d_in[10];
  const float* uni_b   = (const float*)d_in[11];
  float* out = (float*)d_out;

  // workspace carve-out (floats)
  const size_t NH = (size_t)N_ * H_;        // 1,228,800
  float* ws    = (float*)d_ws;
  float* big   = ws; ws += NH;              // raw features  [N,H]
  float* nrm   = ws; ws += NH;              // normalized    [N,H]
  float* intra = ws; ws += (size_t)N_ * L_; // 96 x 64x64 similarity blocks
  float* crossv= ws; ws += (size_t)M_ * M_ * A_;
  float* dinv  = ws; ws += N_;
  float* h0    = ws; ws += NH;              // fc0 output / GCNII h0
  float* hA    = ws; ws += NH;              // GCNII layer-0 output
  float* hBf   = ws; ws += NH;              // GCNII layer-1 output (final h)
  float* hi    = ws; ws += NH;              // adj @ h scratch
  float* hx    = ws; ws += NH;              // 3 x [A,H]
  float* hout  = ws; ws += NH;              // 3 x [A,H] gated outputs

  const int NT = (H_ + 15) / 16;            // 13 N-tiles for H=200

  k_norm <<<N_ / 8,              dim3(256), 0, stream>>>(x_a, x_v, x_t, big, nrm);
  k_gram <<<M_ * B_,             dim3(512), 0, stream>>>(nrm, intra);
  k_cross<<<6 * A_ / 8,          dim3(256), 0, stream>>>(nrm, crossv);
  k_dinv <<<N_ / 8,              dim3(256), 0, stream>>>(intra, crossv, dinv);
  k_fc0  <<<dim3(NT, N_ / 128),  dim3(256), 0, stream>>>(big, fc0_w, fc0_b, h0);

  const float beta0 = logf(0.5f / 1.0f + 1.0f);
  const float beta1 = logf(0.5f / 2.0f + 1.0f);
  k_spmm <<<dim3(NT, M_ * B_),   dim3(128), 0, stream>>>(intra, crossv, dinv, h0, hi);
  k_conv <<<dim3(NT, N_ / 128),  dim3(256), 0, stream>>>(hi, h0, conv_w, beta0, hA);
  k_spmm <<<dim3(NT, M_ * B_),   dim3(128), 0, stream>>>(intra, crossv, dinv, hA, hi);
  k_conv <<<dim3(NT, N_ / 128),  dim3(256), 0, stream>>>(hi, h0, conv_w + 2 * H_ * H_, beta1, hBf);

  k_hx   <<<dim3(NT, A_ / 128, M_), dim3(256), 0, stream>>>(big, hBf, tm_w, tm_b, hx);

  hipMemsetAsync(hout, 0, NH * sizeof(float), stream);
  const int pi_[6] = {0, 0, 1, 1, 2, 2};
  const int pj_[6] = {1, 2, 0, 2, 0, 1};
  for (int p = 0; p < 6; ++p)
    k_gate<<<dim3(NT, A_ / 128), dim3(256), 0, stream>>>(hx, cross_w, cross_b, hout,
                                                         p, pi_[p], pj_[p]);

  k_final<<<A_ / 256, dim3(256), 0, stream>>>(hout, uni_w, uni_b, out);
}